// TinyLSTM_53257594470986
// MI455X (gfx1250) — compile-verified
//
#include <hip/hip_runtime.h>
#include <math.h>

typedef __attribute__((ext_vector_type(2))) float v2f;
typedef __attribute__((ext_vector_type(8))) float v8f;

#define I_SZ 128
#define H_SZ 64
#define G_SZ 256   // 4*H
#define B_SZ 256
#define T_SZ 1024
#define C_SZ 10

// -----------------------------------------------------------------------------
// Kernel 1: x_proj[(t*B + b)*256 + n] = sum_i x[b,t,i]*W_ih[n,i] + b_ih[n]+b_hh[n]
// One wave per 16x16 output tile; K=128 via 32 chained V_WMMA_F32_16X16X4_F32.
// Output stored in (T, B, 4H) order so the scan kernel reads contiguous slabs.
// -----------------------------------------------------------------------------
__global__ __launch_bounds__(256)
void xproj_gemm(const float* __restrict__ x,
                const float* __restrict__ W_ih,
                const float* __restrict__ b_ih,
                const float* __restrict__ b_hh,
                float* __restrict__ xproj) {
    const int lane = threadIdx.x & 31;
    const int wave = threadIdx.x >> 5;
    const int tile = blockIdx.x * 8 + wave;   // 262144 tiles total
    const int mt   = tile >> 4;               // 0..16383  (B*T/16)
    const int nt   = tile & 15;               // 0..15     (4H/16)

    const int rlan = lane & 15;               // A row / B column within tile
    const int kb   = (lane >> 4) * 2;         // K sub-offset per half-wave
    const int half = (lane >> 4) * 8;         // C/D row offset per half-wave

    const float* Abase = x    + (size_t)(mt * 16 + rlan) * I_SZ + kb;
    const float* Bbase = W_ih + (size_t)(nt * 16 + rlan) * I_SZ + kb;

    v8f acc = {};
#pragma unroll 4
    for (int k0 = 0; k0 < I_SZ; k0 += 4) {
        v2f a = *(const v2f*)(Abase + k0);
        v2f b = *(const v2f*)(Bbase + k0);
        acc = __builtin_amdgcn_wmma_f32_16x16x4_f32(
            false, a, false, b, (short)0, acc, false, false);
    }

    const int ncol = nt * 16 + rlan;
    const float bias = b_ih[ncol] + b_hh[ncol];
#pragma unroll
    for (int r = 0; r < 8; ++r) {
        int m = mt * 16 + r + half;           // global row = b*T + t
        int b_idx = m >> 10;                  // / T
        int t_idx = m & (T_SZ - 1);           // % T
        xproj[((size_t)t_idx * B_SZ + b_idx) * G_SZ + ncol] = acc[r] + bias;
    }
}

// -----------------------------------------------------------------------------
// Kernel 2: persistent LSTM scan. 16 blocks x 256 threads (8 waves).
// Block handles batch rows [16*bx, 16*bx+16) for all T steps (rows independent).
// Wave w owns gate columns [32w, 32w+32) -> two 16x16 tiles; W_hh fragments
// live in registers for the whole scan. h lives in LDS; c in thread registers.
// x_proj slab is register double-buffered (load t+1 overlaps compute of t) and
// t+2 is prefetched into cache, so the serial chain never stalls on HBM.
// -----------------------------------------------------------------------------
__global__ __launch_bounds__(256)
void lstm_scan(const float* __restrict__ xproj,
               const float* __restrict__ W_hh,
               const float* __restrict__ W_cls,
               const float* __restrict__ b_cls,
               float* __restrict__ out) {
    __shared__ float hs[16 * 68];    // h, rows padded to 68 floats (bank-safe)
    __shared__ float gs[16 * 260];   // gates, rows padded to 260 floats

    const int tid  = threadIdx.x;
    const int lane = tid & 31;
    const int wave = tid >> 5;
    const int bb   = blockIdx.x * 16;

    const int rlan = lane & 15;
    const int kb   = (lane >> 4) * 2;
    const int half = (lane >> 4) * 8;

    // Preload W_hh B-fragments for this wave's two n-tiles (K=64 -> 16 frags each)
    const int nt0 = wave * 2, nt1 = wave * 2 + 1;
    v2f bf0[16], bf1[16];
#pragma unroll
    for (int ki = 0; ki < 16; ++ki) {
        int k = ki * 4 + kb;
        bf0[ki] = *(const v2f*)(W_hh + (size_t)(nt0 * 16 + rlan) * H_SZ + k);
        bf1[ki] = *(const v2f*)(W_hh + (size_t)(nt1 * 16 + rlan) * H_SZ + k);
    }

    // h = 0
    for (int i = tid; i < 16 * 68; i += 256) hs[i] = 0.0f;

    // Elementwise ownership: thread -> (row, 4 hidden units), c state in regs
    const int erow = tid >> 4;
    const int ejb  = (tid & 15) * 4;
    float c0 = 0.f, c1 = 0.f, c2 = 0.f, c3 = 0.f;

    // Register double-buffer for the x_proj slab: preload step 0
    float xpb0[8], xpb1[8];
    {
        const float* xp0 = xproj + ((size_t)0 * B_SZ + bb) * G_SZ;
#pragma unroll
        for (int r = 0; r < 8; ++r) {
            int row = r + half;
            xpb0[r] = xp0[row * G_SZ + nt0 * 16 + rlan];
            xpb1[r] = xp0[row * G_SZ + nt1 * 16 + rlan];
        }
    }
    __syncthreads();

    for (int t = 0; t < T_SZ; ++t) {
        // A fragments from LDS h (shared by both n-tiles)
        v2f af[16];
#pragma unroll
        for (int ki = 0; ki < 16; ++ki)
            af[ki] = *(const v2f*)(&hs[rlan * 68 + ki * 4 + kb]);

        // Accumulators from the double-buffered x_proj (includes biases)
        v8f acc0, acc1;
#pragma unroll
        for (int r = 0; r < 8; ++r) {
            acc0[r] = xpb0[r];
            acc1[r] = xpb1[r];
        }

        // Issue next step's slab loads now: they overlap the whole WMMA chain,
        // LDS gate exchange, nonlinearities and both barriers of this step.
        if (t + 1 < T_SZ) {
            const float* xpn = xproj + ((size_t)(t + 1) * B_SZ + bb) * G_SZ;
#pragma unroll
            for (int r = 0; r < 8; ++r) {
                int row = r + half;
                xpb0[r] = xpn[row * G_SZ + nt0 * 16 + rlan];
                xpb1[r] = xpn[row * G_SZ + nt1 * 16 + rlan];
            }
        }
        // Prefetch t+2 toward L2/L0 (global_prefetch_b8); 256 thr x 64B = 16KB
        if (t + 2 < T_SZ) {
            const float* xp2 = xproj + ((size_t)(t + 2) * B_SZ + bb) * G_SZ;
            __builtin_prefetch(xp2 + (size_t)tid * 16, 0, 0);
        }

#pragma unroll
        for (int ki = 0; ki < 16; ++ki) {
            acc0 = __builtin_amdgcn_wmma_f32_16x16x4_f32(
                false, af[ki], false, bf0[ki], (short)0, acc0, false, false);
            acc1 = __builtin_amdgcn_wmma_f32_16x16x4_f32(
                false, af[ki], false, bf1[ki], (short)0, acc1, false, false);
        }

        // Spill gate tiles to LDS for the cross-gate elementwise stage
#pragma unroll
        for (int r = 0; r < 8; ++r) {
            int row = r + half;
            gs[row * 260 + nt0 * 16 + rlan] = acc0[r];
            gs[row * 260 + nt1 * 16 + rlan] = acc1[r];
        }
        __syncthreads();

        // Gate nonlinearities + state update; write h_new back to LDS
#pragma unroll
        for (int q = 0; q < 4; ++q) {
            int j = ejb + q;
            float ig = gs[erow * 260 + j];
            float fg = gs[erow * 260 + 64 + j];
            float gg = gs[erow * 260 + 128 + j];
            float og = gs[erow * 260 + 192 + j];
            float it = 1.0f / (1.0f + __expf(-ig));
            float ft = 1.0f / (1.0f + __expf(-fg));
            float gt = tanhf(gg);
            float ot = 1.0f / (1.0f + __expf(-og));
            float& c = (q == 0 ? c0 : q == 1 ? c1 : q == 2 ? c2 : c3);
            c = ft * c + it * gt;
            hs[erow * 68 + j] = ot * tanhf(c);
        }
        __syncthreads();
    }

    // Classifier head: logits[b, cls] = h_T[b,:] . W_cls[cls,:] + b_cls[cls]
    if (tid < 16 * C_SZ) {
        int row = tid / C_SZ, cls = tid % C_SZ;
        float s = b_cls[cls];
#pragma unroll 8
        for (int j = 0; j < H_SZ; ++j)
            s += hs[row * 68 + j] * W_cls[cls * H_SZ + j];
        out[(size_t)(bb + row) * C_SZ + cls] = s;
    }
}

// -----------------------------------------------------------------------------
// Launch: kernel1 (full-chip GEMM) then kernel2 (16 persistent WGs), same stream.
// Workspace: x_proj = 256*1024*256 floats = 256 MB in d_ws.
// -----------------------------------------------------------------------------
extern "C" void kernel_launch(void* const* d_in, const int* in_sizes, int n_in,
                              void* d_out, int out_size, void* d_ws, size_t ws_size,
                              hipStream_t stream) {
    const float* x     = (const float*)d_in[0];
    const float* W_ih  = (const float*)d_in[1];
    const float* W_hh  = (const float*)d_in[2];
    const float* b_ih  = (const float*)d_in[3];
    const float* b_hh  = (const float*)d_in[4];
    const float* W_cls = (const float*)d_in[5];
    const float* b_cls = (const float*)d_in[6];
    float* out   = (float*)d_out;
    float* xproj = (float*)d_ws;   // needs B*T*4H*4 = 256 MB

    // 262144 output tiles / 8 waves per block
    xproj_gemm<<<32768, 256, 0, stream>>>(x, W_ih, b_ih, b_hh, xproj);
    lstm_scan<<<16, 256, 0, stream>>>(xproj, W_hh, W_cls, b_cls, out);
}